// SubgraphRepresentation_73065983640359
// MI455X (gfx1250) — compile-verified
//
#include <hip/hip_runtime.h>
#include <cstdint>

typedef float v2f __attribute__((ext_vector_type(2)));
typedef float v8f __attribute__((ext_vector_type(8)));

__device__ __forceinline__ void atomicMaxFloat(float* addr, float v) {
  if (v >= 0.f) atomicMax((int*)addr, __float_as_int(v));
  else          atomicMin((unsigned int*)addr, __float_as_uint(v));
}

// ---------------------------------------------------------------------------
// GEMM: C[M x 256] = act( A[M x K] @ Wt[K x 256] + bias )
// A may be a concat of two M x 256 buffers (A0 | A1) when K == 512.
// Block = 256 threads (8 waves). Block tile 128(M) x 256(N).
// Wave: 32 rows (2 strips of 16) x 128 cols (8 WMMA n-tiles) -> 16 WMMA / kk.
// B staged in LDS pre-swizzled into WMMA pair layout: (B[k][n],B[k+1][n])
// contiguous -> single ds_load_b64 per operand, no v_mov packing.
// Double-buffered LDS + reg staging pipelines global loads behind compute.
// ---------------------------------------------------------------------------
__global__ __launch_bounds__(256) void gemm_wmma_k(
    const float* __restrict__ A0, const float* __restrict__ A1,
    int K0, int K,
    const float* __restrict__ Wt, const float* __restrict__ bias,
    float* __restrict__ Cout, int M, int do_relu)
{
  // a_s: 128 rows, stride 20 (rows hit distinct banks; 16B-aligned rows)
  // b_s: 8 k-pairs, stride 544 dwords (544%64==32 -> halves on disjoint banks)
  __shared__ __align__(16) float a_s[2][128 * 20];
  __shared__ __align__(16) float b_s[2][8 * 544];

  const int tid   = threadIdx.x;
  const int lane  = tid & 31;
  const int wave  = tid >> 5;
  const int l15   = lane & 15;
  const int hi    = lane >> 4;          // 0: lanes 0-15, 1: lanes 16-31
  const int mpair = wave >> 1;          // 0..3 -> rows mpair*32 .. +32
  const int nhalf = wave & 1;           // 0..1 -> cols nhalf*128 .. +128
  const int blockRow = blockIdx.x * 128;

  v8f acc0[8] = {};
  v8f acc1[8] = {};

  float4 aReg[2];
  float4 bEven[2], bOdd[2];

  const int nch = K >> 4;

  auto loadRegs = [&](int c) {
    const int kc = c << 4;
    // A chunk: 128 rows x 16 cols = 512 float4, 2 per thread
#pragma unroll
    for (int i = 0; i < 2; ++i) {
      int lin = tid + i * 256;
      int r = lin >> 2, seg = lin & 3;
      int grow = blockRow + r;
      float4 v = make_float4(0.f, 0.f, 0.f, 0.f);
      if (grow < M) {
        const float* p;
        if (kc < K0) p = A0 + (size_t)grow * K0 + kc + seg * 4;
        else         p = A1 + (size_t)grow * (K - K0) + (kc - K0) + seg * 4;
        v = *(const float4*)p;
      }
      aReg[i] = v;
    }
    // B chunk: 8 k-pairs x 64 col-quads = 512 slots, 2 per thread
#pragma unroll
    for (int i = 0; i < 2; ++i) {
      int lin = tid + i * 256;
      int p = lin >> 6, nc4 = lin & 63;
      const float* base = Wt + (size_t)(kc + 2 * p) * 256 + nc4 * 4;
      bEven[i] = *(const float4*)base;          // row k = kc+2p
      bOdd[i]  = *(const float4*)(base + 256);  // row k+1
    }
  };

  auto storeLDS = [&](int buf) {
#pragma unroll
    for (int i = 0; i < 2; ++i) {
      int lin = tid + i * 256;
      int r = lin >> 2, seg = lin & 3;
      *(float4*)&a_s[buf][r * 20 + seg * 4] = aReg[i];
    }
#pragma unroll
    for (int i = 0; i < 2; ++i) {
      int lin = tid + i * 256;
      int p = lin >> 6, nc4 = lin & 63;
      float4 e = bEven[i], o = bOdd[i];
      float4 lo = make_float4(e.x, o.x, e.y, o.y);
      float4 hi4 = make_float4(e.z, o.z, e.w, o.w);
      float* dst = &b_s[buf][p * 544 + nc4 * 8];
      *(float4*)dst = lo;
      *(float4*)(dst + 4) = hi4;
    }
  };

  auto computeChunk = [&](int buf) {
#pragma unroll
    for (int kk = 0; kk < 16; kk += 4) {
      const float* ap0 = &a_s[buf][(mpair * 32 + l15) * 20 + kk + 2 * hi];
      v2f a0 = *(const v2f*)ap0;             // (A[m][k], A[m][k+1]) contiguous
      v2f a1 = *(const v2f*)(ap0 + 16 * 20);
      const float* bp = &b_s[buf][((kk >> 1) + hi) * 544 + (nhalf * 128 + l15) * 2];
#pragma unroll
      for (int t = 0; t < 8; ++t) {
        v2f b = *(const v2f*)(bp + t * 32);  // (B[k][n], B[k+1][n]) contiguous
        acc0[t] = __builtin_amdgcn_wmma_f32_16x16x4_f32(
            false, a0, false, b, (short)0, acc0[t], false, false);
        acc1[t] = __builtin_amdgcn_wmma_f32_16x16x4_f32(
            false, a1, false, b, (short)0, acc1[t], false, false);
      }
    }
  };

  loadRegs(0);
  int buf = 0;
  storeLDS(0);
  for (int c = 0; c < nch; ++c) {
    __syncthreads();
    const bool more = (c + 1 < nch);
    if (more) loadRegs(c + 1);   // global loads in flight during compute
    computeChunk(buf);
    if (more) storeLDS(buf ^ 1); // loads have landed by now
    buf ^= 1;
  }

  // epilogue: C/D layout: vgpr r -> row 8*hi + r
#pragma unroll
  for (int t = 0; t < 8; ++t) {
    int n = nhalf * 128 + t * 16 + l15;
    float bv = bias ? bias[n] : 0.f;
#pragma unroll
    for (int r = 0; r < 8; ++r) {
      int row0 = blockRow + mpair * 32 + hi * 8 + r;
      int row1 = row0 + 16;
      if (row0 < M) {
        float v = acc0[t][r] + bv;
        if (do_relu) v = fmaxf(v, 0.f);
        Cout[(size_t)row0 * 256 + n] = v;
      }
      if (row1 < M) {
        float v = acc1[t][r] + bv;
        if (do_relu) v = fmaxf(v, 0.f);
        Cout[(size_t)row1 * 256 + n] = v;
      }
    }
  }
}

// ---------------------------------------------------------------------------
// Small utility kernels
// ---------------------------------------------------------------------------
__global__ void fill_k(float* p, float v, int n) {
  int i = blockIdx.x * blockDim.x + threadIdx.x;
  if (i < n) p[i] = v;
}

// Wt[k*256 + j] = W[j*rowStride + k]; one block (256 threads) per k-row.
__global__ void transpose_k(const float* __restrict__ W, float* __restrict__ Wt,
                            int rowStride) {
  int k = blockIdx.x;
  int j = threadIdx.x;
  Wt[(size_t)k * 256 + j] = W[(size_t)j * rowStride + k];
}

__global__ void qvec_k(const float* __restrict__ qemb, const float* __restrict__ qW,
                       const float* __restrict__ qb, float* __restrict__ q, int D) {
  int j = threadIdx.x;
  float s = qb[j];
  for (int k = 0; k < D; ++k) s += qemb[k] * qW[(size_t)j * D + k];
  q[j] = fmaxf(s, 0.f);
}

// beff[l][j] = nqm_b[l][j] + sum_k q[k] * nqm_W[l][j][256+k]
__global__ void beff_k(const float* __restrict__ nqmW, const float* __restrict__ nqmb,
                       const float* __restrict__ q, float* __restrict__ beff) {
  int l = blockIdx.x, j = threadIdx.x;
  const float* wr = nqmW + ((size_t)l * 256 + j) * 512 + 256;
  float s = nqmb[l * 256 + j];
  for (int k = 0; k < 256; ++k) s += q[k] * wr[k];
  beff[l * 256 + j] = s;
}

// vectorized mix over float4; row = (i4*4)>>8 = i4>>6
__global__ void mix4_k(const float4* __restrict__ x0, const float4* __restrict__ x1,
                       const unsigned char* __restrict__ mask, float4* __restrict__ out,
                       int total4) {
  int i = blockIdx.x * blockDim.x + threadIdx.x;
  if (i >= total4) return;
  float4 a = x0[i], b = x1[i];
  float wa = mask[i >> 6] ? 0.2f : 0.8f;
  float wb = 1.f - wa;
  float4 r;
  r.x = wa * a.x + wb * b.x;
  r.y = wa * a.y + wb * b.y;
  r.z = wa * a.z + wb * b.z;
  r.w = wa * a.w + wb * b.w;
  out[i] = r;
}

// sl[i] = xw[i,:] . asrc ; dl[i] = xw[i,:] . adst  (one wave per row)
__global__ __launch_bounds__(256) void rowdot_k(
    const float* __restrict__ xw, const float* __restrict__ asrc,
    const float* __restrict__ adst, float* __restrict__ sl, float* __restrict__ dl,
    int M) {
  int wave = threadIdx.x >> 5, lane = threadIdx.x & 31;
  int row = blockIdx.x * 8 + wave;
  if (row >= M) return;
  const float* xr = xw + (size_t)row * 256;
  float s = 0.f, d = 0.f;
#pragma unroll
  for (int t = 0; t < 8; ++t) {
    float v = xr[lane + 32 * t];
    s += v * asrc[lane + 32 * t];
    d += v * adst[lane + 32 * t];
  }
#pragma unroll
  for (int off = 16; off; off >>= 1) {
    s += __shfl_down(s, off, 32);
    d += __shfl_down(d, off, 32);
  }
  if (lane == 0) { sl[row] = s; dl[row] = d; }
}

// leaky_relu(sl[s]+dl[d]) -> lg[e]; segment max into m[d] (self-loops appended)
__global__ void logits_k(const int* __restrict__ src, const int* __restrict__ dst,
                         const float* __restrict__ sl, const float* __restrict__ dl,
                         float* __restrict__ lg, float* m, int E, int M) {
  int e = blockIdx.x * blockDim.x + threadIdx.x;
  if (e >= E + M) return;
  int s, d;
  if (e < E) { s = src[e]; d = dst[e]; } else { s = e - E; d = s; }
  float v = sl[s] + dl[d];
  v = v > 0.f ? v : 0.2f * v;
  lg[e] = v;
  atomicMaxFloat(&m[d], v);
}

__global__ void expden_k(const int* __restrict__ dst, const float* lg,
                         const float* __restrict__ m, float* ex, float* den,
                         int E, int M) {
  int e = blockIdx.x * blockDim.x + threadIdx.x;
  if (e >= E + M) return;
  int d = (e < E) ? dst[e] : e - E;
  float v = expf(lg[e] - m[d]);
  ex[e] = v;   // in-place over lg: same element, fine
  atomicAdd(&den[d], v);
}

__global__ void rowbias4_k(const float4* __restrict__ bias4, float4* __restrict__ out,
                           int total4) {
  int i = blockIdx.x * blockDim.x + threadIdx.x;
  if (i < total4) out[i] = bias4[i & 63];
}

// agg[d,:] += (ex[e]/(den[d]+1e-16)) * xw[s,:]   (one wave per edge)
__global__ __launch_bounds__(256) void scatter_k(
    const int* __restrict__ src, const int* __restrict__ dst,
    const float* __restrict__ ex, const float* __restrict__ den,
    const float* __restrict__ xw, float* agg, int E, int M) {
  int wave = threadIdx.x >> 5, lane = threadIdx.x & 31;
  int e = blockIdx.x * 8 + wave;
  if (e >= E + M) return;
  int s, d;
  if (e < E) { s = src[e]; d = dst[e]; } else { s = e - E; d = s; }
  float alpha = ex[e] / (den[d] + 1e-16f);
  const float* xr = xw + (size_t)s * 256;
  float* ar = agg + (size_t)d * 256;
#pragma unroll
  for (int t = 0; t < 8; ++t) {
    int c = lane + 32 * t;
    atomicAdd(&ar[c], alpha * xr[c]);
  }
}

// per-column sum & sum-of-squares over a 128-row strip (thread = column)
__global__ __launch_bounds__(256) void colstats_k(const float* __restrict__ x,
                                                  float* sum, float* sumsq, int M) {
  int j = threadIdx.x;
  int r0 = blockIdx.x * 128;
  int r1 = min(r0 + 128, M);
  float s = 0.f, s2 = 0.f;
  for (int i = r0; i < r1; ++i) {
    float v = x[(size_t)i * 256 + j];
    s += v; s2 += v * v;
  }
  atomicAdd(&sum[j], s);
  atomicAdd(&sumsq[j], s2);
}

// graph-norm as affine: y = scale*x + shift (one-pass variance identity)
__global__ void normcoef_k(const float* __restrict__ sum, const float* __restrict__ sumsq,
                           const float* __restrict__ w, const float* __restrict__ b,
                           const float* __restrict__ a, float* scale, float* shift, int M) {
  int j = threadIdx.x;
  float invM = 1.f / (float)M;
  float mu = sum[j] * invM;
  float am = a[j] * mu;
  float var = sumsq[j] * invM - am * (2.f * mu - am);
  float rinv = rsqrtf(var + 1e-5f);
  scale[j] = w[j] * rinv;
  shift[j] = b[j] - w[j] * rinv * am;
}

__global__ void affine4_k(float4* x, const float4* __restrict__ scale4,
                          const float4* __restrict__ shift4, int total4) {
  int i = blockIdx.x * blockDim.x + threadIdx.x;
  if (i >= total4) return;
  int j = i & 63;
  float4 v = x[i], sc = scale4[j], sh = shift4[j];
  v.x = v.x * sc.x + sh.x;
  v.y = v.y * sc.y + sh.y;
  v.z = v.z * sc.z + sh.z;
  v.w = v.w * sc.w + sh.w;
  x[i] = v;
}

__global__ __launch_bounds__(256) void pool_k(const float* __restrict__ x,
                                              const unsigned char* __restrict__ mask,
                                              float* psum, float* pmax, int M) {
  int j = threadIdx.x;
  int r0 = blockIdx.x * 128;
  int r1 = min(r0 + 128, M);
  float s = 0.f, mx = -INFINITY;
  for (int i = r0; i < r1; ++i) {
    if (mask[i]) {
      float v = x[(size_t)i * 256 + j];
      s += v;
      mx = fmaxf(mx, v);
    }
  }
  atomicAdd(&psum[j], s);
  atomicMaxFloat(&pmax[j], mx);
}

__global__ void count_k(const unsigned char* __restrict__ mask, float* cnt, int M) {
  int i = blockIdx.x * blockDim.x + threadIdx.x;
  int active = (i < M) ? (mask[i] ? 1 : 0) : 0;
  unsigned long long ball = __ballot(active);
  int c = __popcll(ball);
  if ((threadIdx.x & 31) == 0 && c) atomicAdd(cnt, (float)c);
}

// out[j] = final_b[j] + [mean | max | add] . final_W[j,:]
__global__ void final_k(const float* __restrict__ psum, const float* __restrict__ pmax,
                        const float* __restrict__ cnt, const float* __restrict__ fW,
                        const float* __restrict__ fb, float* __restrict__ out) {
  int j = threadIdx.x;
  float c = fmaxf(cnt[0], 1.f);
  const float* wr = fW + (size_t)j * 768;
  float s = fb[j];
  for (int k = 0; k < 256; ++k) s += (psum[k] / c) * wr[k];
  for (int k = 0; k < 256; ++k) s += pmax[k] * wr[256 + k];
  for (int k = 0; k < 256; ++k) s += psum[k] * wr[512 + k];
  out[j] = s;
}

// ---------------------------------------------------------------------------
extern "C" void kernel_launch(void* const* d_in, const int* in_sizes, int n_in,
                              void* d_out, int out_size, void* d_ws, size_t ws_size,
                              hipStream_t stream) {
  const float* x_    = (const float*)d_in[0];
  const int*   eidx  = (const int*)d_in[1];
  // d_in[2] edge_attr: dead in reference (e unused) -> skipped
  const float* qemb  = (const float*)d_in[3];
  const unsigned char* mask = (const unsigned char*)d_in[4];  // jax bool = 1 byte
  const float* node_in_W = (const float*)d_in[5];
  const float* node_in_b = (const float*)d_in[6];
  const float* q_in_W = (const float*)d_in[9];
  const float* q_in_b = (const float*)d_in[10];
  const float* nqm_W  = (const float*)d_in[11];
  const float* nqm_b  = (const float*)d_in[12];
  const float* t0_W   = (const float*)d_in[15];
  const float* t0_b   = (const float*)d_in[16];
  const float* t1_W   = (const float*)d_in[17];
  const float* t1_b   = (const float*)d_in[18];
  const float* gat_W  = (const float*)d_in[19];
  const float* gat_asrc = (const float*)d_in[20];
  const float* gat_adst = (const float*)d_in[21];
  const float* gat_bias = (const float*)d_in[22];
  const float* cgn_w  = (const float*)d_in[23];
  const float* cgn_b  = (const float*)d_in[24];
  const float* cgn_a  = (const float*)d_in[25];
  const float* c0_W   = (const float*)d_in[26];
  const float* c0_b   = (const float*)d_in[27];
  const float* c1_W   = (const float*)d_in[28];
  const float* c1_b   = (const float*)d_in[29];
  const float* gn_w   = (const float*)d_in[30];
  const float* gn_b   = (const float*)d_in[31];
  const float* gn_a   = (const float*)d_in[32];
  const float* final_W = (const float*)d_in[33];
  const float* final_b = (const float*)d_in[34];

  const int N = in_sizes[4];            // 50000
  const int E = in_sizes[1] / 2;        // 400000
  const int D = in_sizes[0] / N;        // 128
  const int* src = eidx;
  const int* dst = eidx + E;

  const size_t NH = (size_t)N * 256;
  float* ws = (float*)d_ws;
  float* XA = ws;             // x
  float* XB = ws + NH;        // xc
  float* XC = ws + 2 * NH;    // x0 / h / agg
  float* XD = ws + 3 * NH;    // x1 / xw
  float* WT = ws + 4 * NH;
  float* wt_node = WT;                                   // D*256
  size_t wtl = (size_t)D * 256;
  const size_t LSTRIDE = 65536ull * 4 + 131072ull * 2;   // nqm,t0,t1,gat + c0,c1
  auto wt_nqm = [&](int l) { return WT + wtl + (size_t)l * LSTRIDE; };
  auto wt_t0  = [&](int l) { return WT + wtl + (size_t)l * LSTRIDE + 65536; };
  auto wt_t1  = [&](int l) { return WT + wtl + (size_t)l * LSTRIDE + 131072; };
  auto wt_gat = [&](int l) { return WT + wtl + (size_t)l * LSTRIDE + 196608; };
  auto wt_c0  = [&](int l) { return WT + wtl + (size_t)l * LSTRIDE + 262144; };
  auto wt_c1  = [&](int l) { return WT + wtl + (size_t)l * LSTRIDE + 393216; };
  float* qv     = WT + wtl + 2 * LSTRIDE;
  float* beff   = qv + 256;          // 2*256
  float* slv    = beff + 512;
  float* dlv    = slv + N;
  float* mv     = dlv + N;
  float* denv   = mv + N;
  float* exv    = denv + N;          // E+N
  float* colsum = exv + (size_t)(E + N);
  float* colsq  = colsum + 256;
  float* scl    = colsq + 256;
  float* shf    = scl + 256;
  float* psum   = shf + 256;
  float* pmax   = psum + 256;
  float* cnt    = pmax + 256;        // 1

  const int total4  = (int)(NH / 4);
  const int gElem4  = (total4 + 255) / 256;
  const int gRows   = (N + 255) / 256;
  const int gEdges  = (E + N + 255) / 256;
  const int gEdgeW  = (E + N + 7) / 8;
  const int gGemm   = (N + 127) / 128;
  const int gStrip  = (N + 127) / 128;
  const int gRowW   = (N + 7) / 8;

  // ---- weight transposes (once per launch) ----
  transpose_k<<<D, 256, 0, stream>>>(node_in_W, wt_node, D);
  for (int l = 0; l < 2; ++l) {
    transpose_k<<<256, 256, 0, stream>>>(nqm_W + (size_t)l * 256 * 512, wt_nqm(l), 512);
    transpose_k<<<256, 256, 0, stream>>>(t0_W + (size_t)l * 65536, wt_t0(l), 256);
    transpose_k<<<256, 256, 0, stream>>>(t1_W + (size_t)l * 65536, wt_t1(l), 256);
    transpose_k<<<256, 256, 0, stream>>>(gat_W + (size_t)l * 65536, wt_gat(l), 256);
    transpose_k<<<512, 256, 0, stream>>>(c0_W + (size_t)l * 256 * 512, wt_c0(l), 512);
    transpose_k<<<512, 256, 0, stream>>>(c1_W + (size_t)l * 256 * 512, wt_c1(l), 512);
  }

  // ---- q and folded nqm bias ----
  qvec_k<<<1, 256, 0, stream>>>(qemb, q_in_W, q_in_b, qv, D);
  beff_k<<<2, 256, 0, stream>>>(nqm_W, nqm_b, qv, beff);

  // ---- input projection: x = relu(x_ @ node_in_W.T + b) ----
  gemm_wmma_k<<<gGemm, 256, 0, stream>>>(x_, nullptr, D, D, wt_node, node_in_b, XA, N, 1);

  for (int l = 0; l < 2; ++l) {
    // xc = relu(x @ nqmA.T + beff)  (q-half of concat folded into beff)
    gemm_wmma_k<<<gGemm, 256, 0, stream>>>(XA, nullptr, 256, 256, wt_nqm(l), beff + l * 256, XB, N, 1);
    // x1 = relu(xc @ t1.T + b); x0 = relu(xc @ t0.T + b)
    gemm_wmma_k<<<gGemm, 256, 0, stream>>>(XB, nullptr, 256, 256, wt_t1(l), t1_b + l * 256, XD, N, 1);
    gemm_wmma_k<<<gGemm, 256, 0, stream>>>(XB, nullptr, 256, 256, wt_t0(l), t0_b + l * 256, XC, N, 1);
    // h = mix(x0, x1)
    mix4_k<<<gElem4, 256, 0, stream>>>((const float4*)XC, (const float4*)XD, mask,
                                       (float4*)XC, total4);
    // xw = h @ gat_W.T
    gemm_wmma_k<<<gGemm, 256, 0, stream>>>(XC, nullptr, 256, 256, wt_gat(l), nullptr, XD, N, 0);
    // GAT attention
    rowdot_k<<<gRowW, 256, 0, stream>>>(XD, gat_asrc + l * 256, gat_adst + l * 256, slv, dlv, N);
    fill_k<<<gRows, 256, 0, stream>>>(mv, -INFINITY, N);
    fill_k<<<gRows, 256, 0, stream>>>(denv, 0.f, N);
    logits_k<<<gEdges, 256, 0, stream>>>(src, dst, slv, dlv, exv, mv, E, N);
    expden_k<<<gEdges, 256, 0, stream>>>(dst, exv, mv, exv, denv, E, N);
    rowbias4_k<<<gElem4, 256, 0, stream>>>((const float4*)(gat_bias + l * 256),
                                           (float4*)XC, total4);  // agg = bias
    scatter_k<<<gEdgeW, 256, 0, stream>>>(src, dst, exv, denv, XD, XC, E, N);
    // graph_norm (cgn) on agg, in place
    fill_k<<<2, 256, 0, stream>>>(colsum, 0.f, 512);
    colstats_k<<<gStrip, 256, 0, stream>>>(XC, colsum, colsq, N);
    normcoef_k<<<1, 256, 0, stream>>>(colsum, colsq, cgn_w + l * 256, cgn_b + l * 256,
                                      cgn_a + l * 256, scl, shf, N);
    affine4_k<<<gElem4, 256, 0, stream>>>((float4*)XC, (const float4*)scl,
                                          (const float4*)shf, total4);
    // hc = [h_norm | xc];  x1 = hc@c1.T+b ; x0 = hc@c0.T+b
    gemm_wmma_k<<<gGemm, 256, 0, stream>>>(XC, XB, 256, 512, wt_c1(l), c1_b + l * 256, XD, N, 0);
    gemm_wmma_k<<<gGemm, 256, 0, stream>>>(XC, XB, 256, 512, wt_c0(l), c0_b + l * 256, XA, N, 0);
    // x = mix(x0, x1); graph_norm (gn) in place
    mix4_k<<<gElem4, 256, 0, stream>>>((const float4*)XA, (const float4*)XD, mask,
                                       (float4*)XA, total4);
    fill_k<<<2, 256, 0, stream>>>(colsum, 0.f, 512);
    colstats_k<<<gStrip, 256, 0, stream>>>(XA, colsum, colsq, N);
    normcoef_k<<<1, 256, 0, stream>>>(colsum, colsq, gn_w + l * 256, gn_b + l * 256,
                                      gn_a + l * 256, scl, shf, N);
    affine4_k<<<gElem4, 256, 0, stream>>>((float4*)XA, (const float4*)scl,
                                          (const float4*)shf, total4);
  }

  // ---- masked pooling + final projection ----
  fill_k<<<1, 256, 0, stream>>>(psum, 0.f, 256);
  fill_k<<<1, 256, 0, stream>>>(pmax, -INFINITY, 256);
  fill_k<<<1, 256, 0, stream>>>(cnt, 0.f, 1);
  pool_k<<<gStrip, 256, 0, stream>>>(XA, mask, psum, pmax, N);
  count_k<<<gRows, 256, 0, stream>>>(mask, cnt, N);
  final_k<<<1, 256, 0, stream>>>(psum, pmax, cnt, final_W, final_b, (float*)d_out);
}